// SplashEncoding_88510686036240
// MI455X (gfx1250) — compile-verified
//
#include <hip/hip_runtime.h>
#include <stdint.h>

#ifndef __has_builtin
#define __has_builtin(x) 0
#endif

#if __has_builtin(__builtin_amdgcn_tensor_load_to_lds)
#define SPLASH_HAVE_TDM 1
#else
#define SPLASH_HAVE_TDM 0
#endif

namespace splash {

constexpr int kLevels = 16;
constexpr int kHM     = 1 << 17;          // hashmap size 2^17
constexpr int kFDim   = 2;

// Replicates the Python _build_cfg() at compile time.
struct Cfg {
  int       res[kLevels];
  int       ns[kLevels];
  int       direct[kLevels];              // res^3 <= HM -> linear index
  long long fb[kLevels + 1];              // feature row offsets
  long long gb[kLevels + 1];              // gaussian row offsets
  constexpr Cfg() : res{}, ns{}, direct{}, fb{}, gb{} {
    double p = 1.0;
    for (int i = 0; i < kLevels; ++i) { res[i] = (int)(16.0 * p); p *= 1.47; }
    for (int i = 0; i < kLevels; ++i) {
      ns[i] = 0;
      if (i >= 15)      ns[i] = 4;        // split 0.9375 -> 16*0.9375 = 15
      else if (i >= 14) ns[i] = 2;        // split 0.875  -> 16*0.875  = 14
    }
    fb[0] = 0; gb[0] = 0;
    for (int i = 0; i < kLevels; ++i) {
      long long r3 = (long long)res[i] * res[i] * res[i];
      direct[i] = (r3 <= kHM) ? 1 : 0;
      long long ni = (r3 < kHM) ? r3 : kHM;
      int s = ns[i] ? ns[i] : 1;
      fb[i + 1] = fb[i] + ni * s;
      gb[i + 1] = gb[i] + ni * (long long)ns[i];
    }
  }
};
constexpr Cfg CFG{};

__device__ __forceinline__ int hash3(int x, int y, int z) {
  // PRIMES = [1, 2654435761, 805459861], xor-combine, mod 2^17
  uint32_t h = (uint32_t)x ^ ((uint32_t)y * 2654435761u) ^ ((uint32_t)z * 805459861u);
  return (int)(h & (uint32_t)(kHM - 1));
}

// Gaussian-splash level (levels 14 & 15, always hashed: res^3 >> 2^17).
template <int LVL, int NSPL>
__device__ __forceinline__ void splash_level(float cx, float cy, float cz,
                                             const float2* __restrict__ feats2,
                                             const float*  __restrict__ means,
                                             const float*  __restrict__ stds,
                                             float* f0, float* f1, float* g) {
  constexpr int res   = CFG.res[LVL];
  constexpr int fbase = (int)CFG.fb[LVL];          // base feature row
  constexpr int gbase = (int)CFG.gb[LVL];          // base gaussian row
  const float resf = (float)res;
  const float hi   = (float)((double)res - 1.001);

  float fx = fminf(fmaxf(resf * cx, 0.0f), hi);
  float fy = fminf(fmaxf(resf * cy, 0.0f), hi);
  float fz = fminf(fmaxf(resf * cz, 0.0f), hi);
  int px = (int)fx, py = (int)fy, pz = (int)fz;
  float tx = fx - (float)px, ty = fy - (float)py, tz = fz - (float)pz;

  float acc0 = 0.0f, acc1 = 0.0f, accg = 0.0f;
#pragma unroll
  for (int k = 0; k < 8; ++k) {
    const int ox = (k >> 2) & 1, oy = (k >> 1) & 1, oz = k & 1;
    float w = (ox ? tx : 1.0f - tx) * (oy ? ty : 1.0f - ty) * (oz ? tz : 1.0f - tz);
    int idx = hash3(px + ox, py + oy, pz + oz);
#pragma unroll
    for (int s = 0; s < NSPL; ++s) {
      int row = idx * NSPL + s;                       // row within level
      float2 f = feats2[(long long)fbase + row];
      const float* m = means + ((long long)gbase + row) * 3;
      float m0 = m[0], m1 = m[1], m2 = m[2];
      float sd = fabsf(stds[(long long)gbase + row]);
      float dx = cx - m0, dy = cy - m1, dz = cz - m2;
      float d2 = dx * dx + dy * dy + dz * dz;
      float gw = expf(-(d2 / (2.0f * sd * sd + 1e-7f))) /
                 (2.5066282746310002f * sd + 1e-7f);   // sqrt(2*pi)
      float inv = 1.0f / sd;
      float dw  = d2 * inv * inv;                      // sum(|diff/s|^2)
      float wg  = w * gw;
      acc0 += wg * f.x;
      acc1 += wg * f.y;
      accg += wg * dw;
    }
  }
  *f0 = acc0; *f1 = acc1; *g = accg;
}

#if SPLASH_HAVE_TDM
typedef unsigned int u32x4 __attribute__((ext_vector_type(4)));
typedef int          i32x8 __attribute__((ext_vector_type(8)));
typedef int          i32x4 __attribute__((ext_vector_type(4)));
#endif

__launch_bounds__(256)
__global__ void splash_encode_kernel(const float* __restrict__ coords,
                                     const float* __restrict__ feats,
                                     const float* __restrict__ means,
                                     const float* __restrict__ stds,
                                     float* __restrict__ out, int N) {
  // Level-0 feature table: 4096 entries * float2 = 32 KB, staged in LDS.
  __shared__ float lds_f0[2 * 4096];

#if SPLASH_HAVE_TDM
  if (threadIdx.x < 32) {                 // wave 0 issues the TDM op (EXEC ignored)
    uint64_t ga   = (uint64_t)(uintptr_t)feats;                 // FB[0] == 0
    uint32_t loff = (uint32_t)(uintptr_t)(void*)lds_f0;         // LDS byte offset
    // D# group 0: count=1 | lds_addr | global_addr[56:0] | type=2
    u32x4 g0 = { 1u, loff, (uint32_t)ga,
                 (uint32_t)((ga >> 32) & 0x01FFFFFFu) | (2u << 30) };
    // D# group 1: data_size=4B; tensor_dim0 = tile_dim0 = 8192 elems; stride0 = 8192
    i32x8 g1 = { (int)(2u << 16),              // data_size = 2 (4 bytes)
                 (int)(8192u << 16),           // tensor_dim0[15:0] @ bits 63:48
                 (int)(1u << 16),              // tensor_dim0 hi=0, tensor_dim1 = 1
                 (int)(8192u << 16),           // tile_dim0 = 8192 @ bits 127:112
                 0,                            // tile_dim1 = tile_dim2 = 0 (unused)
                 8192,                         // tensor_dim0_stride[31:0]
                 (int)(8192u << 16),           // tensor_dim1_stride[15:0] @ 223:208
                 0 };
    i32x4 g2 = { 0, 0, 0, 0 };                 // D# group 2 (unused, <=2D tensor)
    i32x4 g3 = { 0, 0, 0, 0 };                 // D# group 3 (unused, <=2D tensor)
    i32x8 g4 = { 0, 0, 0, 0, 0, 0, 0, 0 };     // extra group (clang-23 6-arg form)
    __builtin_amdgcn_tensor_load_to_lds(g0, g1, g2, g3, g4, 0);
    __builtin_amdgcn_s_wait_tensorcnt(0);
  }
  __syncthreads();
#else
  for (int t = threadIdx.x; t < 2 * 4096; t += blockDim.x) lds_f0[t] = feats[t];
  __syncthreads();
#endif

  int n = blockIdx.x * blockDim.x + threadIdx.x;
  if (n >= N) return;

  float cx = coords[3 * n + 0];
  float cy = coords[3 * n + 1];
  float cz = coords[3 * n + 2];

  float* outf = out + (long long)n * 32;
  float* outg = out + (long long)N * 32 + (long long)n * 2;

  const float2* feats2 = (const float2*)feats;

  // ---- Levels 0..13: plain trilinear hash-grid lookup ----
#pragma unroll
  for (int i = 0; i < 14; ++i) {
    const int res    = CFG.res[i];
    const int direct = CFG.direct[i];
    const float resf = (float)res;
    const float hi   = (float)((double)res - 1.001);

    float fx = fminf(fmaxf(resf * cx, 0.0f), hi);
    float fy = fminf(fmaxf(resf * cy, 0.0f), hi);
    float fz = fminf(fmaxf(resf * cz, 0.0f), hi);
    int px = (int)fx, py = (int)fy, pz = (int)fz;
    float tx = fx - (float)px, ty = fy - (float)py, tz = fz - (float)pz;

    const float2* tab = feats2 + CFG.fb[i];
    float a0 = 0.0f, a1 = 0.0f;
#pragma unroll
    for (int k = 0; k < 8; ++k) {
      const int ox = (k >> 2) & 1, oy = (k >> 1) & 1, oz = k & 1;
      float w = (ox ? tx : 1.0f - tx) * (oy ? ty : 1.0f - ty) *
                (oz ? tz : 1.0f - tz);
      int X = px + ox, Y = py + oy, Z = pz + oz;
      int idx = direct ? (X + Y * res + Z * res * res) : hash3(X, Y, Z);
      float2 f = (i == 0) ? ((const float2*)lds_f0)[idx] : tab[idx];
      a0 += w * f.x;
      a1 += w * f.y;
    }
    ((float2*)outf)[i] = make_float2(a0, a1);
  }

  // ---- Levels 14 & 15: Gaussian splash ----
  float f14a, f14b, g14, f15a, f15b, g15;
  splash_level<14, 2>(cx, cy, cz, feats2, means, stds, &f14a, &f14b, &g14);
  splash_level<15, 4>(cx, cy, cz, feats2, means, stds, &f15a, &f15b, &g15);
  ((float2*)outf)[14] = make_float2(f14a, f14b);
  ((float2*)outf)[15] = make_float2(f15a, f15b);
  ((float2*)outg)[0]  = make_float2(g14, g15);
}

}  // namespace splash

extern "C" void kernel_launch(void* const* d_in, const int* in_sizes, int n_in,
                              void* d_out, int out_size, void* d_ws, size_t ws_size,
                              hipStream_t stream) {
  (void)n_in; (void)out_size; (void)d_ws; (void)ws_size;
  const float* coords = (const float*)d_in[0];
  const float* feats  = (const float*)d_in[1];
  const float* means  = (const float*)d_in[2];
  const float* stds   = (const float*)d_in[3];
  float* out = (float*)d_out;

  int N = in_sizes[0] / 3;                 // coords is (N,3)
  dim3 block(256);
  dim3 grid((N + 255) / 256);
  hipLaunchKernelGGL(splash::splash_encode_kernel, grid, block, 0, stream,
                     coords, feats, means, stds, out, N);
}